// YoloLoss_11141145165897
// MI455X (gfx1250) — compile-verified
//
#include <hip/hip_runtime.h>
#include <hip/hip_bf16.h>
#include <math.h>

// ---------------- problem constants ----------------
#define BB 64
#define SS 52
#define AA 5
#define CC 80
#define TT 20
#define M_TOT (BB * SS * SS * AA)   // 865280
#define M_VEC4 (M_TOT / 4)          // 216320 float4s (exact)
#define N_ITEMS (BB * TT)           // 1280
#define NOOBJ_BLOCKS 512

typedef __attribute__((ext_vector_type(2))) float v2f;
typedef __attribute__((ext_vector_type(8))) float v8f;

// Wave32 sum of per-lane `acc` using V_WMMA_F32_16X16X4_F32.
// A(16x4): a.x -> K=0 (lanes 0-15, M=lane) / K=2 (lanes 16-31, M=lane-16), a.y = 0.
// B(4x16) = ones (layout-independent). D[m][n] = rowsum(m) = acc[m] + acc[m+16].
// Lane L<16 holds D[0..7][L]; lane L>=16 holds D[8..15][L-16].
// Total = local(lane) + local(lane+16) -> valid on lanes 0..15.
__device__ __forceinline__ float wave_sum_wmma(float acc) {
    v2f a; a.x = acc; a.y = 0.0f;
    v2f b; b.x = 1.0f; b.y = 1.0f;
    v8f c = {};
    c = __builtin_amdgcn_wmma_f32_16x16x4_f32(
        /*neg_a=*/false, a, /*neg_b=*/false, b,
        /*c_mod=*/(short)0, c, /*reuse_a=*/false, /*reuse_b=*/false);
    float local = c[0] + c[1] + c[2] + c[3] + c[4] + c[5] + c[6] + c[7];
    return local + __shfl_down(local, 16);   // lane 0 holds the 32-lane total
}

// ---------------- Kernel A: dense noobj partial sums (b128 streaming) ----------------
__global__ void noobj_sum_kernel(const float4* __restrict__ conf4,
                                 float* __restrict__ partials) {
    const int tid = threadIdx.x;
    float acc = 0.0f;
    for (int i = blockIdx.x * blockDim.x + tid; i < M_VEC4; i += gridDim.x * blockDim.x) {
        const float4 p = conf4[i];                 // global_load_b128
        acc += -log1pf(-p.x);
        acc += -log1pf(-p.y);
        acc += -log1pf(-p.z);
        acc += -log1pf(-p.w);
    }
    // wave reduce via WMMA (all lanes reconverged, EXEC == all 1s)
    const float wsum = wave_sum_wmma(acc);
    __shared__ float smem[8];
    const int wid = tid >> 5, lane = tid & 31;
    if (lane == 0) smem[wid] = wsum;
    __syncthreads();
    if (tid == 0) {
        float s = 0.0f;
        #pragma unroll
        for (int w = 0; w < 8; ++w) s += smem[w];
        partials[blockIdx.x] = s;
    }
}

// ---------------- Kernel B: one wave per (b,t) ground-truth item ----------------
// p5 layout (SoA): [0:1280) corr, [1280:2560) obj, [2560:3840) score,
//                  [3840:5120) xy, [5120:6400) wh
__global__ void items_kernel(const float* __restrict__ conf,
                             const float* __restrict__ pred_xy,
                             const float* __restrict__ pred_wh,
                             const float* __restrict__ cls,
                             const float* __restrict__ tlab,
                             const float* __restrict__ tobj,
                             const float* __restrict__ anch,
                             float* __restrict__ p5) {
    const int wid  = threadIdx.x >> 5;
    const int lane = threadIdx.x & 31;
    const int item = blockIdx.x * (blockDim.x >> 5) + wid;   // 0..1279 (exact fit)

    const float inv = 1.0f / 32.0f;
    const float gx = tobj[item * 4 + 0] * inv;
    const float gy = tobj[item * 4 + 1] * inv;
    const float gw = tobj[item * 4 + 2] * inv;
    const float gh = tobj[item * 4 + 3] * inv;
    const int jc = (int)gx;
    const int ic = (int)gy;
    const float garea = gw * gh;

    // anchor IoUs, argmax (first-max tie-break like jnp.argmax)
    float ious[AA];
    float best_iou = -1.0f; int best = 0;
    #pragma unroll
    for (int a = 0; a < AA; ++a) {
        const float aw = anch[a * 2 + 0], ah = anch[a * 2 + 1];
        const float inter = fminf(aw, gw) * fminf(ah, gh);
        const float io = inter / (aw * ah + garea - inter);
        ious[a] = io;
        if (io > best_iou) { best_iou = io; best = a; }
    }

    const size_t cell = (size_t)(item / TT) * (SS * SS) + (size_t)ic * SS + (size_t)jc;
    const size_t slot = cell * AA + (size_t)best;
    const float sw = 2.0f - (gw / (float)SS) * (gh / (float)SS);

    // noobj correction: entries at this cell with mask != 0 (best or ignore)
    float corr = 0.0f;
    if (lane < AA) {
        const bool isb = (lane == best);
        const bool ign = (ious[lane] > 0.6f) && !isb;
        if (isb || ign) corr = -log1pf(-conf[cell * AA + lane]);
    }
    #pragma unroll
    for (int off = 16; off; off >>= 1) corr += __shfl_xor(corr, off);

    // obj BCE at best slot
    const float pw = pred_wh[slot * 2 + 0];
    const float ph = pred_wh[slot * 2 + 1];
    const float inter_p = fminf(pw, gw) * fminf(ph, gh);
    const float iou_p = inter_p / (pw * ph + garea - inter_p);
    const float p = conf[slot];
    const float bce = -(iou_p * logf(p) + (1.0f - iou_p) * log1pf(-p));

    // xy / wh SSE at best slot
    const float px = pred_xy[slot * 2 + 0];
    const float py = pred_xy[slot * 2 + 1];
    const float dxy = (px - gx) * (px - gx) + (py - gy) * (py - gy);
    const float dwh = (pw - gw) * (pw - gw) + (ph - gh) * (ph - gh);

    // cross-entropy: ce = logsumexp(row) - dot(one_hot, row)
    const float* row = cls + slot * CC;
    const float* tl  = tlab + (size_t)item * CC;
    float mx = -3.4e38f;
    for (int c = lane; c < CC; c += 32) mx = fmaxf(mx, row[c]);
    #pragma unroll
    for (int off = 16; off; off >>= 1) mx = fmaxf(mx, __shfl_xor(mx, off));
    float se = 0.0f, dot = 0.0f;
    for (int c = lane; c < CC; c += 32) {
        const float r = row[c];
        se  += expf(r - mx);
        dot += tl[c] * r;
    }
    #pragma unroll
    for (int off = 16; off; off >>= 1) {
        se  += __shfl_xor(se, off);
        dot += __shfl_xor(dot, off);
    }
    const float ce = (logf(se) + mx) - dot;

    if (lane == 0) {
        p5[0 * N_ITEMS + item] = corr;
        p5[1 * N_ITEMS + item] = bce;
        p5[2 * N_ITEMS + item] = sw * ce;
        p5[3 * N_ITEMS + item] = sw * dxy;
        p5[4 * N_ITEMS + item] = sw * dwh;
    }
}

// ---------------- Kernel C: deterministic final reduce + scales ----------------
__device__ float block_reduce_256(float v, float* smem) {
    const int tid = threadIdx.x;
    smem[tid] = v;
    __syncthreads();
    #pragma unroll
    for (int s = 128; s > 0; s >>= 1) {
        if (tid < s) smem[tid] += smem[tid + s];
        __syncthreads();
    }
    const float r = smem[0];
    __syncthreads();
    return r;
}

__global__ void finalize_kernel(const float* __restrict__ pa,   // 512 noobj partials
                                const float* __restrict__ p5,   // 5 x 1280
                                float* __restrict__ out) {
    __shared__ float smem[256];
    const int tid = threadIdx.x;

    float v = 0.0f;
    for (int i = tid; i < NOOBJ_BLOCKS; i += 256) v += pa[i];
    const float s_all = block_reduce_256(v, smem);

    float s5[5];
    #pragma unroll
    for (int k = 0; k < 5; ++k) {
        float u = 0.0f;
        for (int i = tid; i < N_ITEMS; i += 256) u += p5[k * N_ITEMS + i];
        s5[k] = block_reduce_256(u, smem);
    }

    if (tid == 0) {
        const float invB = 1.0f / (float)BB;
        const float noobj = 0.5f * (s_all - s5[0]) * invB;
        const float obj   = 5.0f * s5[1] * invB;
        const float score = 5.0f * s5[2] * invB;
        const float xy    = 5.0f * s5[3] * invB;
        const float wh    = 5.0f * s5[4] * invB;
        out[0] = score;          // total_loss == score_loss
        out[1] = noobj * 0.25f;
        out[2] = obj   * 0.25f;
        out[3] = score * 0.25f;
        out[4] = xy    * 0.25f;
        out[5] = wh    * 0.25f;
    }
}

extern "C" void kernel_launch(void* const* d_in, const int* in_sizes, int n_in,
                              void* d_out, int out_size, void* d_ws, size_t ws_size,
                              hipStream_t stream) {
    const float* conf    = (const float*)d_in[0];
    const float* pred_xy = (const float*)d_in[1];
    const float* pred_wh = (const float*)d_in[2];
    const float* cls     = (const float*)d_in[3];
    const float* tlab    = (const float*)d_in[4];
    const float* tobj    = (const float*)d_in[5];
    const float* anch    = (const float*)d_in[6];
    float* out = (float*)d_out;

    float* ws = (float*)d_ws;
    float* pa = ws;                  // 512 floats
    float* p5 = ws + NOOBJ_BLOCKS;   // 5 * 1280 floats

    noobj_sum_kernel<<<NOOBJ_BLOCKS, 256, 0, stream>>>((const float4*)conf, pa);
    items_kernel<<<N_ITEMS / 8, 256, 0, stream>>>(conf, pred_xy, pred_wh, cls,
                                                  tlab, tobj, anch, p5);
    finalize_kernel<<<1, 256, 0, stream>>>(pa, p5, out);
}